// Equilibrium_Propagation_Value_Network_54296976556048
// MI455X (gfx1250) — compile-verified
//
#include <hip/hip_runtime.h>
#include <math.h>

typedef float v2f __attribute__((ext_vector_type(2)));
typedef float v8f __attribute__((ext_vector_type(8)));
typedef int   v4i __attribute__((ext_vector_type(4)));
typedef __attribute__((address_space(1))) v4i* gptr_b128;   // global (AS1)
typedef __attribute__((address_space(3))) v4i* lptr_b128;   // LDS (AS3)

#define BATCH  16384
#define INPUT  256
#define HIDDEN 512
#define OUTPUT 128
#define LRATE  0.01f
#define BETA1  0.9f
#define BETA2  0.999f
#define EPSA   1e-8f

#if __has_builtin(__builtin_amdgcn_global_load_async_to_lds_b128)
#define HAS_ASYNC_LDS 1
#else
#define HAS_ASYNC_LDS 0
#endif

__device__ __forceinline__ float rho(float s) { return fminf(fmaxf(s, 0.f), 1.f); }
// JAX clip = min(max(x,0),1): lax.max/min split gradient 0.5 at ties.
__device__ __forceinline__ float drho(float s) {
    if (s > 0.f && s < 1.f) return 1.f;
    if (s == 0.f || s == 1.f) return 0.5f;
    return 0.f;
}

// Stage raw 16xK A-tile into LDS (row stride K+4: conflict-free ds_load_b64, 16B rows).
// Uses CDNA5 async global->LDS (no VGPR round-trip, ASYNCcnt) when available.
template <int K>
__device__ __forceinline__ void stage_A(const float* __restrict__ Amat,
                                        float* __restrict__ As, int m_base)
{
    const int tid = threadIdx.x;
    constexpr int LDA = K + 4;
#if HAS_ASYNC_LDS
    for (int i = tid; i < (16 * K) / 4; i += 128) {
        int f = i * 4;
        int r = f / K, c = f % K;
        __builtin_amdgcn_global_load_async_to_lds_b128(
            (gptr_b128)(Amat + (size_t)(m_base + r) * K + c),
            (lptr_b128)(As + r * LDA + c),
            0, 0);
    }
#if __has_builtin(__builtin_amdgcn_s_wait_asynccnt)
    __builtin_amdgcn_s_wait_asynccnt(0);
#else
    asm volatile("s_wait_asynccnt 0x0" ::: "memory");
#endif
#else
    for (int i = tid; i < (16 * K) / 4; i += 128) {
        int f = i * 4;
        int r = f / K, c = f % K;
        *reinterpret_cast<float4*>(As + r * LDA + c) =
            *reinterpret_cast<const float4*>(Amat + (size_t)(m_base + r) * K + c);
    }
#endif
    __syncthreads();
}

// One 16x16 f32 tile of rho(Amat[MxK]) @ B[KxN] via V_WMMA_F32_16X16X4_F32.
// B supplied K-contiguous per column: B[k][n] = BT[n*K + k]  -> one b64 load per WMMA.
// rho() applied at fragment read (clamps co-execute with the XDL WMMA).
template <int K>
__device__ __forceinline__ v8f gemm16(const float* __restrict__ Amat,
                                      const float* __restrict__ BT,
                                      float* __restrict__ As,
                                      int m_base, int n_base)
{
    const int lane = threadIdx.x & 31;
    const int l16  = lane & 15;
    const int koff = (lane >> 4) << 1;   // lanes 0-15 -> K pair {0,1}; 16-31 -> {2,3}
    constexpr int LDA = K + 4;

    stage_A<K>(Amat, As, m_base);

    v8f acc = {0.f, 0.f, 0.f, 0.f, 0.f, 0.f, 0.f, 0.f};
    const float* Bp = BT + (size_t)(n_base + l16) * K + koff;
    const float* Ap = As + l16 * LDA + koff;
#pragma unroll 8
    for (int k = 0; k < K; k += 4) {
        v2f a = *reinterpret_cast<const v2f*>(Ap + k);
        a.x = rho(a.x);
        a.y = rho(a.y);
        v2f b = *reinterpret_cast<const v2f*>(Bp + k);
        // (neg_a, A, neg_b, B, c_mod, C, reuse_a, reuse_b)
        acc = __builtin_amdgcn_wmma_f32_16x16x4_f32(false, a, false, b, (short)0, acc,
                                                    false, false);
    }
    return acc;
}

// Out[MxN] = rho(A) @ B + bias[n]   (precompute A_buf; and Opre each iter)
template <int K, int N>
__global__ __launch_bounds__(128) void k_gemm_bias(const float* __restrict__ Amat,
                                                   const float* __restrict__ BT,
                                                   const float* __restrict__ bias,
                                                   float* __restrict__ Out)
{
    __shared__ __align__(16) float As[16 * (K + 4)];
    const int wave = threadIdx.x >> 5;
    const int lane = threadIdx.x & 31;
    const int l16 = lane & 15, half = lane >> 4;
    const int m_base = blockIdx.y * 16;
    const int n_base = (blockIdx.x * 4 + wave) * 16;
    v8f acc = gemm16<K>(Amat, BT, As, m_base, n_base);
    const int cg = n_base + l16;
    const float bv = bias[cg];
#pragma unroll
    for (int j = 0; j < 8; ++j) {
        int rg = m_base + j + 8 * half;
        Out[(size_t)rg * N + cg] = acc[j] + bv;
    }
}

// Fused: D = rho(o_old) @ W2^T ; pre = A_buf + D ; gh = h - drho(h)*pre ; Adam(h) in place.
// Note W2 itself already has the BT layout for this GEMM: B[k][n] = W2[n*128 + k].
__global__ __launch_bounds__(128) void k_h_update(const float* __restrict__ o_state,
                                                  const float* __restrict__ W2,
                                                  const float* __restrict__ A_buf,
                                                  float* __restrict__ h,
                                                  float* __restrict__ mh,
                                                  float* __restrict__ vh,
                                                  float inv_bc1, float inv_bc2)
{
    constexpr int K = OUTPUT;   // 128
    constexpr int N = HIDDEN;   // 512
    __shared__ __align__(16) float As[16 * (K + 4)];
    const int wave = threadIdx.x >> 5;
    const int lane = threadIdx.x & 31;
    const int l16 = lane & 15, half = lane >> 4;
    const int m_base = blockIdx.y * 16;
    const int n_base = (blockIdx.x * 4 + wave) * 16;
    v8f acc = gemm16<K>(o_state, W2, As, m_base, n_base);
    const int cg = n_base + l16;
#pragma unroll
    for (int j = 0; j < 8; ++j) {
        size_t idx = (size_t)(m_base + j + 8 * half) * N + cg;
        float pre = A_buf[idx] + acc[j];        // b_h folded into A_buf
        float hv  = h[idx];
        float g   = hv - drho(hv) * pre;
        float m   = BETA1 * mh[idx] + (1.f - BETA1) * g;
        float v   = BETA2 * vh[idx] + (1.f - BETA2) * g * g;
        mh[idx] = m; vh[idx] = v;
        h[idx] = hv - LRATE * (m * inv_bc1) / (sqrtf(v * inv_bc2) + EPSA);
    }
}

__global__ __launch_bounds__(256) void k_o_update(float* __restrict__ o,
                                                  float* __restrict__ mo,
                                                  float* __restrict__ vo,
                                                  const float* __restrict__ opre,
                                                  float inv_bc1, float inv_bc2, int n)
{
    int i = blockIdx.x * 256 + threadIdx.x;
    if (i >= n) return;
    float ov = o[i];
    float g  = ov - drho(ov) * opre[i];         // b_o folded into opre
    float m  = BETA1 * mo[i] + (1.f - BETA1) * g;
    float v  = BETA2 * vo[i] + (1.f - BETA2) * g * g;
    mo[i] = m; vo[i] = v;
    o[i] = ov - LRATE * (m * inv_bc1) / (sqrtf(v * inv_bc2) + EPSA);
}

// dst[c*R + r] = src[r*C + c]  (one-off weight transposes into BT layout)
__global__ __launch_bounds__(256) void k_transpose(const float* __restrict__ src,
                                                   float* __restrict__ dst, int R, int C)
{
    int i = blockIdx.x * 256 + threadIdx.x;
    if (i < R * C) {
        int r = i / C, c = i % C;
        dst[c * R + r] = src[i];
    }
}

__global__ __launch_bounds__(256) void k_zero(float4* __restrict__ p, int n4)
{
    int i = blockIdx.x * 256 + threadIdx.x;
    if (i < n4) p[i] = float4{0.f, 0.f, 0.f, 0.f};
}

extern "C" void kernel_launch(void* const* d_in, const int* in_sizes, int n_in,
                              void* d_out, int out_size, void* d_ws, size_t ws_size,
                              hipStream_t stream)
{
    const float* x   = (const float*)d_in[0];   // 16384 x 256
    const float* W1  = (const float*)d_in[1];   // 256 x 512
    const float* W2  = (const float*)d_in[2];   // 512 x 128
    const float* b_h = (const float*)d_in[4];   // 512
    const float* b_o = (const float*)d_in[5];   // 128
    // d_in[3] = b_x (no effect on gradients); d_in[6] = num_iterations = 15 (static)

    const size_t HN = (size_t)BATCH * HIDDEN;   // 8,388,608
    const size_t ON = (size_t)BATCH * OUTPUT;   // 2,097,152

    float* ws    = (float*)d_ws;
    float* A_buf = ws;  ws += HN;                    // rho(x)@W1 + b_h (loop-invariant)
    float* h     = ws;  ws += HN;
    float* mh    = ws;  ws += HN;
    float* vh    = ws;  ws += HN;
    float* Opre  = ws;  ws += ON;
    float* mo    = ws;  ws += ON;
    float* vo    = ws;  ws += ON;
    float* W1T   = ws;  ws += (size_t)INPUT * HIDDEN;   // 512x256: W1T[n*256+k]
    float* W2T   = ws;  ws += (size_t)HIDDEN * OUTPUT;  // 128x512: W2T[n*512+k]
    float* o     = (float*)d_out;

    // Zero-init h/mh/vh/Opre/mo/vo (contiguous) and o.
    {
        size_t nz = 3 * HN + 3 * ON;
        k_zero<<<(int)((nz / 4 + 255) / 256), 256, 0, stream>>>((float4*)h, (int)(nz / 4));
        k_zero<<<(int)((ON / 4 + 255) / 256), 256, 0, stream>>>((float4*)o, (int)(ON / 4));
    }

    // One-off weight transposes into K-contiguous (BT) layout.
    k_transpose<<<(INPUT * HIDDEN + 255) / 256, 256, 0, stream>>>(W1, W1T, INPUT, HIDDEN);
    k_transpose<<<(HIDDEN * OUTPUT + 255) / 256, 256, 0, stream>>>(W2, W2T, HIDDEN, OUTPUT);

    // A_buf = rho(x) @ W1 + b_h   (done once)
    {
        dim3 g(HIDDEN / 64, BATCH / 16);
        k_gemm_bias<INPUT, HIDDEN><<<g, 128, 0, stream>>>(x, W1T, b_h, A_buf);
    }

    for (int t = 1; t <= 15; ++t) {
        float ib1 = (float)(1.0 / (1.0 - pow((double)BETA1, (double)t)));
        float ib2 = (float)(1.0 / (1.0 - pow((double)BETA2, (double)t)));

        // Opre = rho(h_old) @ W2 + b_o      (reads old h; must precede h update)
        {
            dim3 g(OUTPUT / 64, BATCH / 16);
            k_gemm_bias<HIDDEN, OUTPUT><<<g, 128, 0, stream>>>(h, W2T, b_o, Opre);
        }
        // h <- Adam(h, gh) using old o
        {
            dim3 g(HIDDEN / 64, BATCH / 16);
            k_h_update<<<g, 128, 0, stream>>>(o, W2, A_buf, h, mh, vh, ib1, ib2);
        }
        // o <- Adam(o, go) from Opre
        k_o_update<<<(int)((ON + 255) / 256), 256, 0, stream>>>(o, mo, vo, Opre, ib1, ib2, (int)ON);
    }
}